// SparseLoss_17145509446100
// MI455X (gfx1250) — compile-verified
//
#include <hip/hip_runtime.h>

// Problem constants (match the reference).
#define B_N 4096
#define C_N 32
#define D_N 2048
#define K_N 128

// ---- monotone float <-> uint key mapping (total order, ascending) ----------
__device__ __forceinline__ unsigned fkey(float x) {
    unsigned u = __float_as_uint(x);
    return u ^ ((u & 0x80000000u) ? 0xFFFFFFFFu : 0x80000000u);
}
__device__ __forceinline__ float funkey(unsigned k) {
    unsigned u = (k & 0x80000000u) ? (k ^ 0x80000000u) : ~k;
    return __uint_as_float(u);
}

// ---------------------------------------------------------------------------
// Kernel A: precompute lp = log_softmax(rho) tables (128 values).
// ws layout (floats): [0..127]=lp, [128..255]=log(-lp), [256..383]=1-lp,
//                     [384..511]=log(1-lp)
// ---------------------------------------------------------------------------
__global__ void lp_prep_kernel(const float* __restrict__ rho,
                               float* __restrict__ ws) {
    __shared__ float red[K_N];
    const int t = threadIdx.x;          // 0..127
    float v = rho[t];

    red[t] = v; __syncthreads();
    for (int s = 64; s > 0; s >>= 1) {
        if (t < s) red[t] = fmaxf(red[t], red[t + s]);
        __syncthreads();
    }
    float m = red[0]; __syncthreads();

    red[t] = expf(v - m); __syncthreads();
    for (int s = 64; s > 0; s >>= 1) {
        if (t < s) red[t] += red[t + s];
        __syncthreads();
    }
    float lse = m + logf(red[0]);

    float lp = v - lse;                 // strictly negative
    float a2 = 1.0f - lp;               // > 1
    ws[t]          = lp;
    ws[K_N + t]    = logf(-lp);
    ws[2*K_N + t]  = a2;
    ws[3*K_N + t]  = logf(a2);
}

// ---------------------------------------------------------------------------
// Kernel B: one block per sample. Async-copy the labeled row into LDS,
// radix-select + sort the 128 smallest, logsumexp, KL terms -> partial[b].
// ---------------------------------------------------------------------------
__global__ __launch_bounds__(256) void row_select_kernel(
        const float* __restrict__ encoded,
        const int*   __restrict__ labels,
        const float* __restrict__ lpws,
        float*       __restrict__ partial) {

    __shared__ __align__(16) float s_row[D_N];   // 8 KB gathered row
    __shared__ unsigned s_hist[256];             // histogram -> in-place scan
    __shared__ float    s_sel[K_N];
    __shared__ float    s_red[256];
    __shared__ unsigned s_bcast[2];              // [0]=bin/counter, [1]=rank

    const int tid = threadIdx.x;
    const int b   = blockIdx.x;
    const int lab = labels[b];
    const float* rowp = encoded + ((size_t)b * C_N + (size_t)lab) * (size_t)D_N;

    // ---- CDNA5 async global -> LDS copy (GLOBAL_LOAD_ASYNC_TO_LDS_B128) ----
    // 256 lanes x 16B x 2 passes = 8 KB. Low 32 bits of a flat shared address
    // are the LDS byte offset (ISA 10.2 aperture mapping).
    #pragma unroll
    for (int p = 0; p < 2; ++p) {
        const int off = (p * 256 + tid) * 4;     // float index, 16B aligned
        unsigned lds_addr = (unsigned)(unsigned long long)(&s_row[off]);
        unsigned long long gaddr = (unsigned long long)(rowp + off);
        asm volatile("global_load_async_to_lds_b128 %0, %1, off"
                     :: "v"(lds_addr), "v"(gaddr) : "memory");
    }
    asm volatile("s_wait_asynccnt 0" ::: "memory");   // wave's ASYNCcnt -> 0
    __syncthreads();                                  // cross-wave visibility

    // ---- exact 4-pass byte radix select: key of rank 127 (0-indexed) ------
    unsigned prefix = 0;
    unsigned kth    = K_N - 1;
    #pragma unroll
    for (int r = 0; r < 4; ++r) {
        const int shift = 24 - 8 * r;
        const unsigned mask = (r == 0) ? 0u : (0xFFFFFFFFu << (shift + 8));

        s_hist[tid] = 0;
        __syncthreads();
        for (int i = tid; i < D_N; i += 256) {
            unsigned k = fkey(s_row[i]);
            if (((k ^ prefix) & mask) == 0)
                atomicAdd(&s_hist[(k >> shift) & 0xFF], 1u);
        }
        __syncthreads();

        // Parallel in-place inclusive scan over the 256 bins (Hillis-Steele).
        const unsigned myh = s_hist[tid];
        #pragma unroll
        for (int d = 1; d < 256; d <<= 1) {
            unsigned v = (tid >= d) ? s_hist[tid - d] : 0u;
            __syncthreads();
            s_hist[tid] += v;
            __syncthreads();
        }
        const unsigned incl = s_hist[tid];
        const unsigned lo   = incl - myh;      // exclusive prefix
        // Exactly one bin's [lo, incl) range contains the target rank.
        if (kth >= lo && kth < incl) {
            s_bcast[0] = (unsigned)tid;        // chosen bin
            s_bcast[1] = kth - lo;             // rank within the chosen bin
        }
        __syncthreads();
        prefix |= (s_bcast[0] << shift);
        kth     = s_bcast[1];
        __syncthreads();
    }
    const unsigned T = prefix;          // exact key of the 128th smallest

    // ---- compact keys < T, pad ties with value(T) -------------------------
    if (tid == 0) s_bcast[0] = 0;
    __syncthreads();
    for (int i = tid; i < D_N; i += 256) {
        float x = s_row[i];
        if (fkey(x) < T) {
            unsigned p = atomicAdd(&s_bcast[0], 1u);
            s_sel[p] = x;
        }
    }
    __syncthreads();
    const unsigned nLess = s_bcast[0];  // <= 127 by construction
    const float tv = funkey(T);
    for (unsigned i = nLess + (unsigned)tid; i < K_N; i += 256) s_sel[i] = tv;
    __syncthreads();

    // ---- bitonic sort of 128 floats, ascending ----------------------------
    for (unsigned sz = 2; sz <= K_N; sz <<= 1) {
        for (unsigned j = sz >> 1; j > 0; j >>= 1) {
            __syncthreads();
            if (tid < K_N) {
                unsigned ixj = (unsigned)tid ^ j;
                if (ixj > (unsigned)tid) {
                    float a = s_sel[tid];
                    float c = s_sel[ixj];
                    bool up = (((unsigned)tid & sz) == 0);
                    if ((a > c) == up) { s_sel[tid] = c; s_sel[ixj] = a; }
                }
            }
        }
    }
    __syncthreads();

    // ---- logsumexp over the 128 selected values ---------------------------
    const float vmax = s_sel[K_N - 1];  // ascending -> last is max
    float e = (tid < K_N) ? expf(s_sel[tid] - vmax) : 0.0f;
    s_red[tid] = e; __syncthreads();
    for (int s = 128; s > 0; s >>= 1) {
        if (tid < s) s_red[tid] += s_red[tid + s];
        __syncthreads();
    }
    const float lse = vmax + logf(s_red[0]);
    __syncthreads();

    // ---- KL terms ---------------------------------------------------------
    float term = 0.0f;
    if (tid < K_N) {
        float lq  = s_sel[tid] - lse;    // strictly negative
        float nlq = -lq;                 // positive
        float a1 = lpws[tid];
        float c1 = lpws[K_N + tid];
        float a2 = lpws[2*K_N + tid];
        float c2 = lpws[3*K_N + tid];
        // lp*log(lp/lq) = a1*(log(-lp) - log(-lq));  1-lq = 1 + nlq
        term = a1 * (c1 - logf(nlq)) + a2 * (c2 - logf(1.0f + nlq));
    }
    s_red[tid] = term; __syncthreads();
    for (int s = 128; s > 0; s >>= 1) {
        if (tid < s) s_red[tid] += s_red[tid + s];
        __syncthreads();
    }
    if (tid == 0) partial[b] = s_red[0];
}

// ---------------------------------------------------------------------------
// Kernel C: deterministic fixed-order reduction of the 4096 partials.
// ---------------------------------------------------------------------------
__global__ void final_reduce_kernel(const float* __restrict__ partial,
                                    float* __restrict__ out) {
    __shared__ float s_red[256];
    const int t = threadIdx.x;
    float acc = 0.0f;
    for (int i = t; i < B_N; i += 256) acc += partial[i];
    s_red[t] = acc; __syncthreads();
    for (int s = 128; s > 0; s >>= 1) {
        if (t < s) s_red[t] += s_red[t + s];
        __syncthreads();
    }
    if (t == 0) out[0] = s_red[0];
}

// ---------------------------------------------------------------------------
extern "C" void kernel_launch(void* const* d_in, const int* in_sizes, int n_in,
                              void* d_out, int out_size, void* d_ws, size_t ws_size,
                              hipStream_t stream) {
    const float* rho     = (const float*)d_in[0];   // [1,128]
    const float* encoded = (const float*)d_in[1];   // [4096,32,2048]
    const int*   labels  = (const int*)d_in[2];     // [4096]
    // d_in[3] = K (always 128, hardcoded)

    float* ws      = (float*)d_ws;
    float* lpws    = ws;            // 512 floats: lp tables
    float* partial = ws + 4 * K_N;  // 4096 floats: per-row partial sums

    lp_prep_kernel  <<<1,   K_N, 0, stream>>>(rho, lpws);
    row_select_kernel<<<B_N, 256, 0, stream>>>(encoded, labels, lpws, partial);
    final_reduce_kernel<<<1, 256, 0, stream>>>(partial, (float*)d_out);
}